// QuatizationEmbedding_26654567039199
// MI455X (gfx1250) — compile-verified
//
#include <hip/hip_runtime.h>

// Product-quantization embedding gather for MI455X (gfx1250).
// Pure data-movement op: store-bandwidth bound (~164 MB output @ 23.3 TB/s).
//   - tables (23 MB total) stay resident in the 192 MB L2 via regular-temporal loads
//   - write-once output uses non-temporal B128 stores (512 contiguous B per wave-store)
//   - 16 lanes cooperate per (batch,field) row; 4 independent gather chains per
//     thread hide the 3-deep dependent-load latency (x -> index -> codebook)
//   - chain pair-stride 159,744 = 39*4096 is ≡ 0 (mod 39), so the field id and all
//     field-derived offsets are chain-invariant and computed ONCE per thread.

typedef __attribute__((ext_vector_type(4))) float f32x4;

namespace {
constexpr int kFields        = 39;
constexpr int kFieldDim      = 10000;          // all FIELD_DIMS are 10000
constexpr int kMaxK          = 1024;
constexpr int kEmbed         = 64;
constexpr int kM             = 8;              // sub-spaces, PLEN = 8 floats each
constexpr int kBatch         = 16384;
constexpr int kGrain         = 4;              // independent chains per thread
constexpr int kSlotsTotal    = kBatch * kFields * (kEmbed / 4); // 10,223,616 float4 slots
constexpr int kSlotsPerChain = kSlotsTotal / kGrain;            // 2,555,904
constexpr int kPairsPerChain = kSlotsPerChain / 16;             // 159,744 = 39 * 4096
constexpr int kThreads       = 256;            // 8 waves (wave32) per block
constexpr int kBlocks        = kSlotsPerChain / kThreads;       // 9,984 (exact)

static_assert(kPairsPerChain % kFields == 0, "field id must be chain-invariant");
static_assert(kSlotsPerChain % 16 == 0, "sub-slot must be chain-invariant");
} // namespace

__global__ __launch_bounds__(kThreads)
void qembed_gather_kernel(const int* __restrict__ x,
                          const int* __restrict__ cbw,   // cb_index_weight (390000 x 8)
                          const float* __restrict__ cb,  // codebooks (39936 x 64)
                          float* __restrict__ out)       // (16384 x 39 x 64)
{
    const int gid = blockIdx.x * kThreads + threadIdx.x;

    // Chain-invariant decomposition (computed once per thread).
    const int sub   = gid & 15;        // which float4 of the 64-float output row
    const int m     = sub >> 1;        // sub-space 0..7
    const int col   = (sub & 1) * 4 + m * 8;   // column within the codebook row
    const int pair0 = gid >> 4;        // flat (batch, field) id for chain 0
    const int f     = pair0 % kFields; // invariant: chain stride ≡ 0 (mod 39)
    const int xoff  = f * kFieldDim;   // cumulative field offset into cb_index_weight
    const int roff  = f * kMaxK;       // field offset into codebooks

#pragma unroll
    for (int k = 0; k < kGrain; ++k) {
        const int pair = pair0 + k * kPairsPerChain;

        // x -> sub-index -> codebook row (dependent gather chain, 4 in flight).
        const int xo  = x[pair] + xoff;
        const int row = cbw[xo * kM + m] + roff;

        // Gather 4 floats: codebooks[row, col .. col+3]. Table is L2-hot.
        const f32x4 v = *(const f32x4*)(cb + (row * kEmbed + col));

        // Output float4 slot: pair*16 + sub. Non-temporal: write-once stream
        // must not evict the gather tables from L2.
        __builtin_nontemporal_store(v, (f32x4*)(out + ((size_t)pair * 16 + sub) * 4));
    }
}

extern "C" void kernel_launch(void* const* d_in, const int* in_sizes, int n_in,
                              void* d_out, int out_size, void* d_ws, size_t ws_size,
                              hipStream_t stream) {
    const int*   x   = (const int*)d_in[0];   // (16384, 39) int32
    const int*   cbw = (const int*)d_in[1];   // (390000, 8) int32
    const float* cb  = (const float*)d_in[2]; // (39936, 64) fp32
    float*       out = (float*)d_out;         // (16384, 39, 64) fp32

    qembed_gather_kernel<<<kBlocks, kThreads, 0, stream>>>(x, cbw, cb, out);
}